// SpatialTransformer_60705067762009
// MI455X (gfx1250) — compile-verified
//
#include <hip/hip_runtime.h>

// Volume geometry from the reference: vol [B, 192, 192, 192, 1] fp32.
#define VDIM 192

// Output tile per workgroup: 32 (w, fastest) x 8 (h) x 16 (d).
// Block is (32, 8, 4): each thread computes 4 d-slices (stride blockDim.z).
#define TW_TILE 32
#define TH_TILE 8
#define TD_TILE 16
#define DZ      4
#define SMEM_FLOATS 10240   // 40 KB LDS staging buffer for the input bbox

#if defined(__gfx1250__) && __has_builtin(__builtin_amdgcn_tensor_load_to_lds)
#define USE_TDM 1
typedef unsigned int tdm_v4u __attribute__((ext_vector_type(4)));
typedef int          tdm_v4i __attribute__((ext_vector_type(4)));
typedef int          tdm_v8i __attribute__((ext_vector_type(8)));
#endif

// Per-axis corner indices / weights with the reference's exact clamp rules:
//   cl = clip(l, 0, 191); f0 = clip(floor(l), 0, 191); f1 = clip(f0+1, 0, 191)
//   w0 = f1 - cl (floor corner), w1 = 1 - w0 (ceil corner)
__device__ __forceinline__ void axis_prep(float l, int& i0, int& i1,
                                          float& w0, float& w1) {
    const float MX = (float)(VDIM - 1);
    float cl = fminf(fmaxf(l, 0.0f), MX);
    float f0 = fminf(fmaxf(floorf(l), 0.0f), MX);
    float f1 = fminf(f0 + 1.0f, MX);
    w0 = f1 - cl;
    w1 = 1.0f - w0;
    i0 = (int)f0;
    i1 = (int)f1;
}

__global__ __launch_bounds__(TW_TILE * TH_TILE * DZ) void warp_trilinear_tdm(
    const float* __restrict__ vol,   // [B,192,192,192]
    const float* __restrict__ aff,   // [B,3,4]
    float* __restrict__ out)         // [B,192,192,192]
{
    __shared__ float smem[SMEM_FLOATS];

    const int tw0 = blockIdx.x * TW_TILE;
    const int th0 = blockIdx.y * TH_TILE;
    const int ndt = VDIM / TD_TILE;              // 12 d-tiles
    const int td0 = (blockIdx.z % ndt) * TD_TILE;
    const int b   = blockIdx.z / ndt;

    // Block-uniform affine coefficients -> scalar loads / scalar f32 math.
    const float* A = aff + b * 12;
    const float a00 = A[0], a01 = A[1], a02 = A[2],  a03 = A[3];
    const float a10 = A[4], a11 = A[5], a12 = A[6],  a13 = A[7];
    const float a20 = A[8], a21 = A[9], a22 = A[10], a23 = A[11];

    const float C = 0.5f * (float)(VDIM - 1);    // 95.5

    // ---- Exact input bounding box of this tile (linear map + independent
    // coords => range = value(center) +/- sum(|coef| * half_extent)). ----
    const float hd = 0.5f * (float)(TD_TILE - 1);
    const float hh = 0.5f * (float)(TH_TILE - 1);
    const float hw = 0.5f * (float)(TW_TILE - 1);
    const float cdm = (float)td0 + hd - C;
    const float chm = (float)th0 + hh - C;
    const float cwm = (float)tw0 + hw - C;

    int bmin[3], dims[3];
    {
        const float r0c[3] = {a00, a10, a20};
        const float r1c[3] = {a01, a11, a21};
        const float r2c[3] = {a02, a12, a22};
        const float r3c[3] = {a03, a13, a23};
#pragma unroll
        for (int ax = 0; ax < 3; ++ax) {
            float vc  = r0c[ax] * cdm + r1c[ax] * chm + r2c[ax] * cwm + r3c[ax] + C;
            float rad = fabsf(r0c[ax]) * hd + fabsf(r1c[ax]) * hh + fabsf(r2c[ax]) * hw;
            int lo = (int)floorf(vc - rad);
            int hi = (int)floorf(vc + rad) + 1;
            lo = max(0, min(lo, VDIM - 1));
            hi = max(0, min(hi, VDIM - 1));
            bmin[ax] = lo;
            dims[ax] = hi - lo + 1;
        }
    }
    const int dimD = dims[0], dimH = dims[1], dimW = dims[2];

    bool use_lds = false;
#ifdef USE_TDM
    use_lds = (dimD * dimH * dimW) <= SMEM_FLOATS;
    // Wave 0 of the block (tx 0..31, ty==0, tz==0) issues one TDM descriptor
    // for the whole tile; TDM ignores EXEC and signals TENSORcnt once done.
    if (use_lds && threadIdx.y == 0 && threadIdx.z == 0) {
        const unsigned long long gaddr = (unsigned long long)(const void*)vol +
            4ull * (unsigned long long)(((b * VDIM + bmin[0]) * VDIM + bmin[1]) * VDIM + bmin[2]);
        const unsigned lds_base = (unsigned)(unsigned long long)(void*)smem;

        // D# group 0: count=1 | lds_addr | global_addr[56:0] | type=2
        tdm_v4u g0;
        g0.x = 1u;
        g0.y = lds_base;
        g0.z = (unsigned)(gaddr & 0xffffffffull);
        g0.w = (unsigned)((gaddr >> 32) & 0x1ffffffull) | (2u << 30);

        // D# group 1: data_size=4B; tensor dims 192x192; tile dims = bbox
        // (dim0 = W fastest); strides 192 / 192*192 elements.
        const unsigned td0_ = (unsigned)VDIM;            // tensor_dim0
        const unsigned td1_ = (unsigned)VDIM;            // tensor_dim1
        const unsigned s0_  = (unsigned)VDIM;            // tensor_dim0_stride
        const unsigned s1_  = (unsigned)(VDIM * VDIM);   // tensor_dim1_stride
        tdm_v8i g1;
        g1[0] = (int)(2u << 16);                                   // data_size=2 (4B)
        g1[1] = (int)((td0_ & 0xffffu) << 16);                     // tensor_dim0[15:0]
        g1[2] = (int)((td0_ >> 16) | ((td1_ & 0xffffu) << 16));    // td0[31:16]|td1[15:0]
        g1[3] = (int)((td1_ >> 16) | ((unsigned)dimW << 16));      // td1[31:16]|tile_dim0
        g1[4] = (int)((unsigned)dimH | ((unsigned)dimD << 16));    // tile_dim1|tile_dim2
        g1[5] = (int)s0_;                                          // stride0[31:0]
        g1[6] = (int)(((s0_ >> 32) & 0u) | ((s1_ & 0xffffu) << 16)); // stride0[47:32]|stride1[15:0]
        g1[7] = (int)(s1_ >> 16);                                  // stride1[47:16]

        // D# group 2: tensor_dim2=192; dim3 unused; stride2 = 192^3.
        tdm_v4i g2;
        g2[0] = VDIM;
        g2[1] = 0;
        g2[2] = VDIM * VDIM * VDIM;
        g2[3] = 0;

        tdm_v4i g3 = {0, 0, 0, 0};

#if __clang_major__ >= 23
        tdm_v8i g4 = {0, 0, 0, 0, 0, 0, 0, 0};
        __builtin_amdgcn_tensor_load_to_lds(g0, g1, g2, g3, g4, 0);
#else
        __builtin_amdgcn_tensor_load_to_lds(g0, g1, g2, g3, 0);
#endif
#if __has_builtin(__builtin_amdgcn_s_wait_tensorcnt)
        __builtin_amdgcn_s_wait_tensorcnt(0);
#endif
    }
#endif // USE_TDM
    __syncthreads();   // LDS tile visible to all 32 waves

    const int w = tw0 + threadIdx.x;
    const int h = th0 + threadIdx.y;
    const float hcf = (float)h - C;
    const float wcf = (float)w - C;
    const float* vb = vol + b * (VDIM * VDIM * VDIM);

#pragma unroll
    for (int k = 0; k < TD_TILE / DZ; ++k) {
        const int d = td0 + threadIdx.z + k * DZ;
        const float dcf = (float)d - C;

        const float ld = a00 * dcf + a01 * hcf + a02 * wcf + a03 + C;
        const float lh = a10 * dcf + a11 * hcf + a12 * wcf + a13 + C;
        const float lw = a20 * dcf + a21 * hcf + a22 * wcf + a23 + C;

        int id0, id1, ih0, ih1, iw0, iw1;
        float wd0, wd1, wh0, wh1, ww0, ww1;
        axis_prep(ld, id0, id1, wd0, wd1);
        axis_prep(lh, ih0, ih1, wh0, wh1);
        axis_prep(lw, iw0, iw1, ww0, ww1);

        float res;
        if (use_lds) {
            // All clamped corner indices provably lie inside the bbox
            // (clamp/floor are monotone, bbox built from exact axis ranges).
            const int zd0 = id0 - bmin[0], zd1 = id1 - bmin[0];
            const int yh0 = ih0 - bmin[1], yh1 = ih1 - bmin[1];
            const int xw0 = iw0 - bmin[2], xw1 = iw1 - bmin[2];
            const int p00 = (zd0 * dimH + yh0) * dimW;
            const int p01 = (zd0 * dimH + yh1) * dimW;
            const int p10 = (zd1 * dimH + yh0) * dimW;
            const int p11 = (zd1 * dimH + yh1) * dimW;
            float r00 = smem[p00 + xw0] * ww0 + smem[p00 + xw1] * ww1;
            float r01 = smem[p01 + xw0] * ww0 + smem[p01 + xw1] * ww1;
            float r10 = smem[p10 + xw0] * ww0 + smem[p10 + xw1] * ww1;
            float r11 = smem[p11 + xw0] * ww0 + smem[p11 + xw1] * ww1;
            res = (r00 * wh0 + r01 * wh1) * wd0 + (r10 * wh0 + r11 * wh1) * wd1;
        } else {
            // Uniform fallback: direct global gather (correct for any affine).
            const int q00 = (id0 * VDIM + ih0) * VDIM;
            const int q01 = (id0 * VDIM + ih1) * VDIM;
            const int q10 = (id1 * VDIM + ih0) * VDIM;
            const int q11 = (id1 * VDIM + ih1) * VDIM;
            float r00 = vb[q00 + iw0] * ww0 + vb[q00 + iw1] * ww1;
            float r01 = vb[q01 + iw0] * ww0 + vb[q01 + iw1] * ww1;
            float r10 = vb[q10 + iw0] * ww0 + vb[q10 + iw1] * ww1;
            float r11 = vb[q11 + iw0] * ww0 + vb[q11 + iw1] * ww1;
            res = (r00 * wh0 + r01 * wh1) * wd0 + (r10 * wh0 + r11 * wh1) * wd1;
        }

        // Write-once output: NT store keeps the 192MB L2 for the input volume.
        const int oidx = ((b * VDIM + d) * VDIM + h) * VDIM + w;
        __builtin_nontemporal_store(res, out + oidx);
    }
}

extern "C" void kernel_launch(void* const* d_in, const int* in_sizes, int n_in,
                              void* d_out, int out_size, void* d_ws, size_t ws_size,
                              hipStream_t stream) {
    const float* vol = (const float*)d_in[0];   // [B,192,192,192,1] fp32
    const float* aff = (const float*)d_in[1];   // [B,3,4] fp32
    float* out = (float*)d_out;

    const int B = (n_in > 1 && in_sizes[1] >= 12) ? (in_sizes[1] / 12) : 2;

    dim3 block(TW_TILE, TH_TILE, DZ);                        // 1024 thr = 32 waves
    dim3 grid(VDIM / TW_TILE, VDIM / TH_TILE,
              (VDIM / TD_TILE) * B);                         // (6, 24, 12*B)
    warp_trilinear_tdm<<<grid, block, 0, stream>>>(vol, aff, out);
}